// SvegaSlowMaxPool2d_5248450036553
// MI455X (gfx1250) — compile-verified
//
#include <hip/hip_runtime.h>

// MaxPool2d kernel=2 stride=2, VALID, NCHW.
// Input  X: (16, 64, 512, 512) fp32  -> treated as 1024 images of 512x512
// Output Y: (16, 64, 256, 256) fp32  -> 1024 images of 256x256
//
// Pure streaming op: 1.25 GiB traffic, ~57us floor at 23.3 TB/s.
// Strategy: b128 loads/stores with non-temporal (TH_NT) cache hints,
// 4 outputs per thread (one float4 store), shift-only index math.

typedef float v4f __attribute__((ext_vector_type(4)));

__global__ __launch_bounds__(256) void
SvegaSlowMaxPool2d_5248450036553_kernel(const float* __restrict__ X,
                                        float* __restrict__ Y,
                                        int nQuads) {
    int q = blockIdx.x * blockDim.x + threadIdx.x;
    if (q >= nQuads) return;

    // 16384 output float4-quads per image: 256 rows x 64 quads/row
    int img = q >> 14;
    int rem = q & 16383;
    int h   = rem >> 6;   // output row (input rows 2h, 2h+1)
    int wq  = rem & 63;   // quad within output row (8 input floats)

    // Base index in float4 units: img*65536 + (2h)*128 + wq*2
    long base = ((long)img << 16) + ((long)h << 8) + ((long)wq << 1);

    const v4f* in4 = (const v4f*)X;

    // Two float4s from each of the two input rows (row pitch = 128 float4s)
    v4f a0 = __builtin_nontemporal_load(in4 + base);
    v4f a1 = __builtin_nontemporal_load(in4 + base + 1);
    v4f b0 = __builtin_nontemporal_load(in4 + base + 128);
    v4f b1 = __builtin_nontemporal_load(in4 + base + 129);

    // Vertical max (elementwise), then horizontal pairwise max
    v4f m0, m1;
    m0.x = fmaxf(a0.x, b0.x);
    m0.y = fmaxf(a0.y, b0.y);
    m0.z = fmaxf(a0.z, b0.z);
    m0.w = fmaxf(a0.w, b0.w);
    m1.x = fmaxf(a1.x, b1.x);
    m1.y = fmaxf(a1.y, b1.y);
    m1.z = fmaxf(a1.z, b1.z);
    m1.w = fmaxf(a1.w, b1.w);

    v4f r;
    r.x = fmaxf(m0.x, m0.y);
    r.y = fmaxf(m0.z, m0.w);
    r.z = fmaxf(m1.x, m1.y);
    r.w = fmaxf(m1.z, m1.w);

    __builtin_nontemporal_store(r, (v4f*)Y + q);
}

extern "C" void kernel_launch(void* const* d_in, const int* in_sizes, int n_in,
                              void* d_out, int out_size, void* d_ws, size_t ws_size,
                              hipStream_t stream) {
    const float* X = (const float*)d_in[0];
    float*       Y = (float*)d_out;

    // out_size = 16*64*256*256 = 67,108,864 elements; 4 per thread
    int nQuads = out_size / 4;             // 16,777,216
    const int block = 256;                 // 8 wave32 waves
    int grid = (nQuads + block - 1) / block;  // 65,536 blocks

    SvegaSlowMaxPool2d_5248450036553_kernel<<<grid, block, 0, stream>>>(X, Y, nQuads);
}